// RBFN_ode_25211458027950
// MI455X (gfx1250) — compile-verified
//
#include <hip/hip_runtime.h>

// Fused RBFN forward for MI455X (gfx1250, wave32):
//   dist2 = ||y||^2 - 2*(Y @ C^T) + ||c||^2        (GEMM1 via v_wmma_f32_16x16x4_f32)
//   phi   = exp(-dist2 * exp(-2*log_sigma))
//   out   = phi @ W^T + b                          (GEMM2 via v_wmma_f32_16x16x4_f32)
// Phi never touches HBM: each 16x16 phi tile lives in a per-wave LDS buffer.

typedef float v2f __attribute__((ext_vector_type(2)));
typedef float v8f __attribute__((ext_vector_type(8)));

#if defined(__HIP_DEVICE_COMPILE__) && !__has_builtin(__builtin_amdgcn_wmma_f32_16x16x4_f32)
#error "gfx1250: __builtin_amdgcn_wmma_f32_16x16x4_f32 not available on this toolchain"
#endif

#define NN   8192
#define CC   512
#define DD   64
#define OO   64
#define WAVES 4
#define ROWS_PER_BLOCK (WAVES * 16)   // 64
#define NBLOCKS (NN / ROWS_PER_BLOCK) // 128

// LDS layout (floats): sC[512][64] | sW[64][512] | sCn[512] | sG[512] | sYn[4][16] | sPhi[4][256]
#define SM_C    0
#define SM_W    (SM_C  + CC * DD)
#define SM_CN   (SM_W  + OO * CC)
#define SM_G    (SM_CN + CC)
#define SM_YN   (SM_G  + CC)
#define SM_PHI  (SM_YN + WAVES * 16)
#define SM_TOTAL_FLOATS (SM_PHI + WAVES * 256)

__global__ __launch_bounds__(WAVES * 32)
void rbfn_fused_wmma(const float* __restrict__ y,
                     const float* __restrict__ centres,
                     const float* __restrict__ log_sigmas,
                     const float* __restrict__ W,
                     const float* __restrict__ b,
                     float* __restrict__ out)
{
    extern __shared__ float smem[];
    float* sC   = smem + SM_C;    // centres, row-major [c][d]
    float* sW   = smem + SM_W;    // W, row-major [o][c]
    float* sCn  = smem + SM_CN;   // ||c||^2
    float* sG   = smem + SM_G;    // exp(-2*log_sigma)
    float* sYn  = smem + SM_YN;   // per-wave row norms
    float* sPhi = smem + SM_PHI;  // per-wave 16x16 phi tile

    const int tid = threadIdx.x;
    const int nthreads = WAVES * 32;

    // ---- Block-cooperative staging (float4 vectorized copies) ----
    {
        const float4* src = (const float4*)centres;
        float4*       dst = (float4*)sC;
        for (int i = tid; i < (CC * DD) / 4; i += nthreads) dst[i] = src[i];
        const float4* srw = (const float4*)W;
        float4*       dsw = (float4*)sW;
        for (int i = tid; i < (OO * CC) / 4; i += nthreads) dsw[i] = srw[i];
    }
    for (int c = tid; c < CC; c += nthreads) {
        const float4* cr = (const float4*)(centres + c * DD);
        float s = 0.f;
        #pragma unroll
        for (int k = 0; k < DD / 4; ++k) {
            float4 v = cr[k];
            s = fmaf(v.x, v.x, fmaf(v.y, v.y, fmaf(v.z, v.z, fmaf(v.w, v.w, s))));
        }
        sCn[c] = s;
        sG[c]  = __expf(-2.f * log_sigmas[c]);
    }
    __syncthreads();

    // ---- Per-wave: one 16-row tile of N ----
    const int wave = tid >> 5;
    const int lane = tid & 31;
    const int h    = lane >> 4;   // K-half select (A/B fragment lane-half)
    const int lm   = lane & 15;   // M (A) or N (B) index within tile
    const int m0   = blockIdx.x * ROWS_PER_BLOCK + wave * 16;

    // A fragments of the Y tile: frag k covers K=4k..4k+3.
    // Lane l: m = l%16; V0 = Y[m][4k+2h], V1 = Y[m][4k+2h+1]  -> contiguous b64 load.
    const float* yrow = y + (size_t)(m0 + lm) * DD;
    v2f a[16];
    #pragma unroll
    for (int k = 0; k < 16; ++k)
        a[k] = *(const v2f*)(yrow + 4 * k + 2 * h);

    // Row norms from the fragments: lane l holds half of row (l%16); pair-sum via xor-16 shuffle.
    float pn = 0.f;
    #pragma unroll
    for (int k = 0; k < 16; ++k)
        pn = fmaf(a[k].x, a[k].x, fmaf(a[k].y, a[k].y, pn));
    pn += __shfl_xor(pn, 16);
    if (h == 0) sYn[wave * 16 + lm] = pn;   // same-wave DS ops are in-order; no barrier needed
    float yn[8];
    #pragma unroll
    for (int r = 0; r < 8; ++r) yn[r] = sYn[wave * 16 + r + 8 * h];

    const v8f zero8 = {0.f, 0.f, 0.f, 0.f, 0.f, 0.f, 0.f, 0.f};
    v8f acc[4];
    #pragma unroll
    for (int t = 0; t < 4; ++t) acc[t] = zero8;

    float* myPhi = sPhi + wave * 256;

    for (int j = 0; j < CC / 16; ++j) {       // 32 column tiles of C
        // ---- GEMM1: S = Y * C^T for this 16x16 tile (K=64 -> 16 wmma, 2 acc chains) ----
        v8f s0 = zero8, s1 = zero8;
        const float* crow = sC + (size_t)(16 * j + lm) * DD;  // B[k][n] = centres[n][k]
        #pragma unroll
        for (int k = 0; k < 16; k += 2) {
            v2f bf0 = *(const v2f*)(crow + 4 * k + 2 * h);
            v2f bf1 = *(const v2f*)(crow + 4 * (k + 1) + 2 * h);
            s0 = __builtin_amdgcn_wmma_f32_16x16x4_f32(false, a[k],     false, bf0, (short)0, s0, false, false);
            s1 = __builtin_amdgcn_wmma_f32_16x16x4_f32(false, a[k + 1], false, bf1, (short)0, s1, false, false);
        }

        // ---- Epilogue: phi = exp(-(||y||^2 - 2S + ||c||^2) * g) -> per-wave LDS tile ----
        const int   ng = 16 * j + lm;         // D-frag: element (m = r + 8h, n = lm)
        const float cn = sCn[ng];
        const float gg = sG[ng];
        #pragma unroll
        for (int r = 0; r < 8; ++r) {
            float sv = s0[r] + s1[r];
            float d2 = fmaxf(yn[r] - 2.f * sv + cn, 0.f);
            myPhi[(r + 8 * h) * 16 + lm] = __expf(-d2 * gg);
        }

        // ---- GEMM2 partial: out += Phi[:,16j:16j+16] * W^T chunk (K=16 -> 4 frags x 4 out tiles) ----
        #pragma unroll
        for (int kk = 0; kk < 4; ++kk) {
            v2f pa = *(const v2f*)(myPhi + lm * 16 + 4 * kk + 2 * h);   // A frag from phi tile
            #pragma unroll
            for (int t = 0; t < 4; ++t) {
                // B[k][n] = W[n][k], n = 16t+lm, k = 16j+4kk+2h (+1) -> contiguous b64 in sW
                v2f wb = *(const v2f*)(sW + (size_t)(16 * t + lm) * CC + 16 * j + 4 * kk + 2 * h);
                acc[t] = __builtin_amdgcn_wmma_f32_16x16x4_f32(false, pa, false, wb, (short)0, acc[t], false, false);
            }
        }
    }

    // ---- Bias + store: element (m = r + 8h, o = 16t + lm) ----
    #pragma unroll
    for (int t = 0; t < 4; ++t) {
        float bv = b[16 * t + lm];
        #pragma unroll
        for (int r = 0; r < 8; ++r)
            out[(size_t)(m0 + r + 8 * h) * OO + 16 * t + lm] = acc[t][r] + bv;
    }
}

extern "C" void kernel_launch(void* const* d_in, const int* in_sizes, int n_in,
                              void* d_out, int out_size, void* d_ws, size_t ws_size,
                              hipStream_t stream) {
    (void)in_sizes; (void)n_in; (void)d_ws; (void)ws_size; (void)out_size;
    const float* y  = (const float*)d_in[0];
    const float* ce = (const float*)d_in[1];
    const float* ls = (const float*)d_in[2];
    const float* W  = (const float*)d_in[3];
    const float* b  = (const float*)d_in[4];
    // d_in[5] is t (unused by forward pass)
    float* out = (float*)d_out;

    constexpr size_t smem_bytes = SM_TOTAL_FLOATS * sizeof(float);  // ~264 KB < 320 KB/WGP
    dim3 grid(NBLOCKS), block(WAVES * 32);
    hipLaunchKernelGGL(rbfn_fused_wmma, grid, block, smem_bytes, stream,
                       y, ce, ls, W, b, out);
}